// LIGHTGCN_Extractor_30872224923929
// MI455X (gfx1250) — compile-verified
//
#include <hip/hip_runtime.h>
#include <stdint.h>

// ---------------- problem constants (match reference) ----------------
#define STU    100000
#define EXER   50000
#define KNOW   123
#define DIM    64
#define NN     (STU + EXER)       // 150000 nodes
#define NE     1200000            // edges
#define BATCH  8192
#define LAYERS 3

#define EPB    64                 // edges staged per block
#define WPB    8                  // wave32s per 256-thread block

typedef uint32_t u32;
typedef uint64_t u64;
typedef __attribute__((ext_vector_type(4))) u32 u32x4;
typedef __attribute__((ext_vector_type(8))) int i32x8;
typedef __attribute__((ext_vector_type(4))) int i32x4;

// ---------------- TDM: async 1-D tile load Global -> LDS --------------
// Builds a Tensor DMA Descriptor (D#) per CDNA5 ISA ch.8 and issues
// tensor_load_to_lds (tracked by TENSORcnt).
//   group0: [1:0]count=1 | [63:32]lds_addr | [120:64]global_addr | [127:126]type=2
//   group1: [17:16]data_size=2(4B) | [79:48]tensor_dim0 | [111:80]tensor_dim1=1
//           | [127:112]tile_dim0 | [207:160]tensor_dim0_stride
__device__ __forceinline__ void tdm_load_1d(u32 lds_off, const void* gptr,
                                            u32 tile_elems, u32 tensor_elems) {
#if __has_builtin(__builtin_amdgcn_tensor_load_to_lds)
  u64 ga = (u64)gptr;
  u32x4 g0;
  g0.s0 = 1u;                                            // count=1, user mode
  g0.s1 = lds_off;                                       // lds_addr (bytes)
  g0.s2 = (u32)ga;                                       // global_addr[31:0]
  g0.s3 = (u32)((ga >> 32) & 0x01FFFFFFu) | (2u << 30);  // addr[56:32] | type=2
  i32x8 g1;
  g1.s0 = (int)(2u << 16);                               // data_size = 4 bytes
  g1.s1 = (int)((tensor_elems & 0xFFFFu) << 16);         // tensor_dim0[15:0]
  g1.s2 = (int)(((tensor_elems >> 16) & 0xFFFFu)         // tensor_dim0[31:16]
                | (1u << 16));                           // tensor_dim1 = 1
  g1.s3 = (int)((tile_elems & 0xFFFFu) << 16);           // tile_dim0
  g1.s4 = 0;                                             // tile_dim1/2 = 0 (1-D)
  g1.s5 = (int)tensor_elems;                             // tensor_dim0_stride lo
  g1.s6 = 0;
  g1.s7 = 0;
  i32x4 z4 = {0, 0, 0, 0};                               // groups 2/3 unused (<=2D)
  i32x8 z8 = {0, 0, 0, 0, 0, 0, 0, 0};
  // 6-arg form (clang-23 / therock HIP headers on this toolchain)
  __builtin_amdgcn_tensor_load_to_lds(g0, g1, z4, z4, z8, 0);
#else
  (void)lds_off; (void)gptr; (void)tile_elems; (void)tensor_elems;
#endif
}

__device__ __forceinline__ void tdm_wait0() {
#if __has_builtin(__builtin_amdgcn_s_wait_tensorcnt)
  __builtin_amdgcn_s_wait_tensorcnt(0);
#endif
}

// ---------------- f32 global atomic add (want global_atomic_add_f32) --
__device__ __forceinline__ void fadd_atomic(float* p, float v) {
#if __has_builtin(__builtin_amdgcn_global_atomic_fadd_f32)
  __builtin_amdgcn_global_atomic_fadd_f32(
      (__attribute__((address_space(1))) float*)(uintptr_t)p, v);
#else
  atomicAdd(p, v);
#endif
}

// ---------------- kernels --------------------------------------------
// One pass of msgs = val * h[col]; h_next = segment_sum(msgs, row).
// Block stages 64 edges' metadata into LDS via TDM; each wave32 owns one
// edge per iteration, each lane a float2 slice of the 64-float row.
// (First function in the file so the disasm snippet shows the hot kernel.)
__global__ void __launch_bounds__(256)
k_scatter(const int* __restrict__ erow,
          const int* __restrict__ ecol,
          const float* __restrict__ eval,
          const float* __restrict__ hin,
          float* __restrict__ hout) {
  __shared__ int   s_row[EPB];
  __shared__ int   s_col[EPB];
  __shared__ float s_val[EPB];
  const u32 ebase = blockIdx.x * EPB;

#if __has_builtin(__builtin_amdgcn_tensor_load_to_lds)
  if (threadIdx.x < 32) {                    // wave 0 issues the DMA
    tdm_load_1d((u32)(u64)(void*)s_row, erow + ebase, EPB, EPB);
    tdm_load_1d((u32)(u64)(void*)s_col, ecol + ebase, EPB, EPB);
    tdm_load_1d((u32)(u64)(void*)s_val, eval + ebase, EPB, EPB);
    tdm_wait0();
  }
#else
  if (threadIdx.x < EPB) {                   // fallback: plain staging
    s_row[threadIdx.x] = erow[ebase + threadIdx.x];
    s_col[threadIdx.x] = ecol[ebase + threadIdx.x];
    s_val[threadIdx.x] = eval[ebase + threadIdx.x];
  }
#endif
  __syncthreads();

  const int wave = threadIdx.x >> 5;
  const int lane = threadIdx.x & 31;
#pragma unroll
  for (int it = 0; it < EPB / WPB; ++it) {
    const int   el  = wave + it * WPB;
    const int   row = s_row[el];
    const int   col = s_col[el];
    const float v   = s_val[el];
    const float2 x  = *((const float2*)(hin + (size_t)col * DIM) + lane);
    float* dst = hout + (size_t)row * DIM + (size_t)lane * 2;
    fadd_atomic(dst,     v * x.x);
    fadd_atomic(dst + 1, v * x.y);
  }
}

// h_cur = concat(stu_emb, exer_emb), float4-vectorized
__global__ void __launch_bounds__(256)
k_init_h(const float4* __restrict__ stu,
         const float4* __restrict__ exer,
         float4* __restrict__ h) {
  int i = blockIdx.x * blockDim.x + threadIdx.x;         // over NN*16
  if (i >= NN * 16) return;
  h[i] = (i < STU * 16) ? stu[i] : exer[i - STU * 16];
}

__global__ void __launch_bounds__(256)
k_zero(float4* __restrict__ h) {
  int i = blockIdx.x * blockDim.x + threadIdx.x;
  if (i < NN * 16) h[i] = make_float4(0.f, 0.f, 0.f, 0.f);
}

// layer-0 contribution of acc: out_s = stu_emb[sid], out_d = exer_emb[eid]
__global__ void __launch_bounds__(256)
k_gather_init(const int* __restrict__ sid,
              const int* __restrict__ eid,
              const float4* __restrict__ stu,
              const float4* __restrict__ exer,
              float4* __restrict__ out) {
  int t = blockIdx.x * blockDim.x + threadIdx.x;         // BATCH*16
  if (t >= BATCH * 16) return;
  int i = t >> 4, c = t & 15;
  out[t]              = stu[(size_t)sid[i] * 16 + c];
  out[BATCH * 16 + t] = exer[(size_t)eid[i] * 16 + c];
}

// accumulate this layer's h into the gathered outputs
__global__ void __launch_bounds__(256)
k_gather_add(const int* __restrict__ sid,
             const int* __restrict__ eid,
             const float4* __restrict__ h,
             float4* __restrict__ out) {
  int t = blockIdx.x * blockDim.x + threadIdx.x;         // BATCH*16
  if (t >= BATCH * 16) return;
  int i = t >> 4, c = t & 15;
  float4 a = out[t];
  float4 b = h[(size_t)sid[i] * 16 + c];
  a.x += b.x; a.y += b.y; a.z += b.z; a.w += b.w;
  out[t] = a;
  float4 a2 = out[BATCH * 16 + t];
  float4 b2 = h[((size_t)STU + (size_t)eid[i]) * 16 + c];
  a2.x += b2.x; a2.y += b2.y; a2.z += b2.z; a2.w += b2.w;
  out[BATCH * 16 + t] = a2;
}

// disc_ts = disc_emb[eid]; knowledge_ts = know_emb (copies)
__global__ void __launch_bounds__(256)
k_tail(const int* __restrict__ eid,
       const float* __restrict__ disc,
       const float* __restrict__ know,
       float* __restrict__ out) {
  int t = blockIdx.x * blockDim.x + threadIdx.x;
  if (t < BATCH)
    out[2 * BATCH * DIM + t] = disc[eid[t]];
  else if (t < BATCH + KNOW * DIM)
    out[2 * BATCH * DIM + t] = know[t - BATCH];
}

// light_out = acc / (GCN_LAYERS+1) on the gathered regions
__global__ void __launch_bounds__(256)
k_scale(float4* __restrict__ out) {
  int t = blockIdx.x * blockDim.x + threadIdx.x;         // 2*BATCH*16
  if (t >= 2 * BATCH * 16) return;
  float4 a = out[t];
  a.x *= 0.25f; a.y *= 0.25f; a.z *= 0.25f; a.w *= 0.25f;
  out[t] = a;
}

// ---------------- host launcher --------------------------------------
extern "C" void kernel_launch(void* const* d_in, const int* in_sizes, int n_in,
                              void* d_out, int out_size, void* d_ws, size_t ws_size,
                              hipStream_t stream) {
  (void)in_sizes; (void)n_in; (void)out_size;
  const int*   sid  = (const int*)d_in[0];
  const int*   eid  = (const int*)d_in[1];
  /* d_in[2] q_mask unused by the reference outputs */
  const int*   erow = (const int*)d_in[3];
  const int*   ecol = (const int*)d_in[4];
  const float* evl  = (const float*)d_in[5];
  const float* stu  = (const float*)d_in[6];
  const float* exer = (const float*)d_in[7];
  const float* disc = (const float*)d_in[8];
  const float* know = (const float*)d_in[9];
  float* out = (float*)d_out;

  // ping-pong node feature buffers in workspace (2 * 150000 * 64 f32 = 76.8 MB)
  if (ws_size < 2ull * NN * DIM * sizeof(float)) return;
  float* h0 = (float*)d_ws;
  float* h1 = h0 + (size_t)NN * DIM;

  const dim3 blk(256);
  const int g_nodes  = (NN * 16 + 255) / 256;
  const int g_batch  = (BATCH * 16 + 255) / 256;

  k_init_h<<<g_nodes, blk, 0, stream>>>((const float4*)stu, (const float4*)exer,
                                        (float4*)h0);
  k_gather_init<<<g_batch, blk, 0, stream>>>(sid, eid, (const float4*)stu,
                                             (const float4*)exer, (float4*)out);
  k_tail<<<(BATCH + KNOW * DIM + 255) / 256, blk, 0, stream>>>(eid, disc, know, out);

  float* hc = h0;
  float* hn = h1;
  for (int l = 0; l < LAYERS; ++l) {
    k_zero<<<g_nodes, blk, 0, stream>>>((float4*)hn);
    k_scatter<<<NE / EPB, blk, 0, stream>>>(erow, ecol, evl, hc, hn);
    k_gather_add<<<g_batch, blk, 0, stream>>>(sid, eid, (const float4*)hn,
                                              (float4*)out);
    float* t = hc; hc = hn; hn = t;
  }
  k_scale<<<(2 * BATCH * 16 + 255) / 256, blk, 0, stream>>>((float4*)out);
}